// OnlineTripleLoss_84396107366993
// MI455X (gfx1250) — compile-verified
//
#include <hip/hip_runtime.h>

typedef float v2f __attribute__((ext_vector_type(2)));
typedef float v8f __attribute__((ext_vector_type(8)));

#define BSZ   8192
#define DIM   128
#define TILE  128
#define LDSR  132          // padded row stride (floats); 528 B row = 33*16, keeps b128/b64 alignment
#define TMARGIN 0.2f

__device__ __forceinline__ float half16_min(float v) {
    // min across each 16-lane half independently (xor masks < 16 stay in-half)
    #pragma unroll
    for (int off = 1; off < 16; off <<= 1)
        v = fminf(v, __shfl_xor(v, off, 32));
    return v;
}

// CDNA5 direct global->LDS async copy (GVS mode): INST_OFFSET is added to BOTH
// the per-lane LDS address and the per-lane global offset, so one (lds, voff)
// pair + literal offsets covers a contiguous half-row. Tracked by ASYNCcnt.
template <int OFF>
__device__ __forceinline__ void async_b128(unsigned lds, int voff, const float* base) {
    asm volatile("global_load_async_to_lds_b128 %0, %1, %2 offset:%c3"
                 :: "v"(lds), "v"(voff), "s"(base), "n"(OFF) : "memory");
}

__device__ __forceinline__ void async_half_row(unsigned lds, int voff, const float* base) {
    async_b128<0>(lds, voff, base);   async_b128<16>(lds, voff, base);
    async_b128<32>(lds, voff, base);  async_b128<48>(lds, voff, base);
    async_b128<64>(lds, voff, base);  async_b128<80>(lds, voff, base);
    async_b128<96>(lds, voff, base);  async_b128<112>(lds, voff, base);
    async_b128<128>(lds, voff, base); async_b128<144>(lds, voff, base);
    async_b128<160>(lds, voff, base); async_b128<176>(lds, voff, base);
    async_b128<192>(lds, voff, base); async_b128<208>(lds, voff, base);
    async_b128<224>(lds, voff, base); async_b128<240>(lds, voff, base);
}

// Kernel 1: per-row squared norms, init min_an = +inf, zero accumulators.
__global__ __launch_bounds__(256)
void sq_init_kernel(const float* __restrict__ emb, float* __restrict__ sq,
                    float* __restrict__ min_an, float* __restrict__ acc) {
    if (blockIdx.x == 0 && threadIdx.x == 0) {
        acc[0] = 0.0f;
        ((unsigned int*)acc)[1] = 0u;
    }
    int lane = threadIdx.x & 31;
    int wave = threadIdx.x >> 5;
    int row  = blockIdx.x * 8 + wave;
    if (row < BSZ) {
        const float4* p = (const float4*)(emb + (size_t)row * DIM);
        float4 x = p[lane];                       // 32 lanes x float4 = 128 floats
        float s = x.x * x.x + x.y * x.y + x.z * x.z + x.w * x.w;
        #pragma unroll
        for (int off = 1; off < 32; off <<= 1) s += __shfl_xor(s, off, 32);
        if (lane == 0) {
            sq[row]     = s;
            min_an[row] = __builtin_inff();
        }
    }
}

// MODE 0: mining pass (row-wise min dist over different-label cols -> min_an)
// MODE 1: loss pass  (sum/count of relu(dist + margin - min_an[row]) over valid pairs)
template <int MODE>
__global__ __launch_bounds__(256)
void tile_kernel(const float* __restrict__ emb, const int* __restrict__ labels,
                 const float* __restrict__ sq, float* __restrict__ min_an,
                 float* __restrict__ acc_out) {
    const int rowBase = blockIdx.y * TILE;
    const int colBase = blockIdx.x * TILE;

    // Pass 2 only needs row < col: skip tiles strictly below the diagonal.
    if (MODE == 1 && (colBase + TILE) <= rowBase) return;

    __shared__ float As[TILE * LDSR];
    __shared__ float Bs[TILE * LDSR];

    // Async fill: thread t owns half a row (64 floats = 16 x b128) of each tile.
    {
        int r = threadIdx.x >> 1;
        int h = (threadIdx.x & 1) * 64;
        unsigned ldsA = (unsigned)(size_t)(As + r * LDSR + h);
        unsigned ldsB = (unsigned)(size_t)(Bs + r * LDSR + h);
        int voffA = ((rowBase + r) * DIM + h) * 4;   // byte offset from emb
        int voffB = ((colBase + r) * DIM + h) * 4;
        async_half_row(ldsA, voffA, emb);
        async_half_row(ldsB, voffB, emb);
        asm volatile("s_wait_asynccnt 0" ::: "memory");
    }
    __syncthreads();

    const int lane = threadIdx.x & 31;
    const int wave = threadIdx.x >> 5;   // 8 waves: wave w -> tile rows [16w, 16w+16)
    const int l15  = lane & 15;
    const int hi   = lane >> 4;          // k-half select for A/B fragments

    v8f acc[8];
    #pragma unroll
    for (int c = 0; c < 8; ++c) acc[c] = (v8f){0.f,0.f,0.f,0.f,0.f,0.f,0.f,0.f};

    // A frag: lane -> row (16w + l15), VGPR pair = K (k+2*hi, k+2*hi+1). B mirrors with n=l15.
    const float* aRow = As + (16 * wave + l15) * LDSR + 2 * hi;
    const float* bRow = Bs + l15 * LDSR + 2 * hi;

    #pragma unroll 4
    for (int k = 0; k < DIM; k += 4) {
        v2f a = *(const v2f*)(aRow + k);
        #pragma unroll
        for (int c = 0; c < 8; ++c) {
            v2f b = *(const v2f*)(bRow + c * 16 * LDSR + k);
            acc[c] = __builtin_amdgcn_wmma_f32_16x16x4_f32(
                false, a, false, b, (short)0, acc[c], false, false);
        }
    }

    // C/D layout: VGPR r, lanes 0-15 -> (M=r, N=lane); lanes 16-31 -> (M=r+8, N=lane-16).
    const int rowW = rowBase + 16 * wave + 8 * hi;   // + r gives the global row
    float sqc[8]; int labc[8];
    #pragma unroll
    for (int c = 0; c < 8; ++c) {
        int col = colBase + 16 * c + l15;
        sqc[c]  = sq[col];
        labc[c] = labels[col];
    }
    float sqr[8], mar[8]; int labr[8];
    #pragma unroll
    for (int r = 0; r < 8; ++r) {
        int row = rowW + r;
        sqr[r]  = sq[row];
        labr[r] = labels[row];
        if (MODE == 1) mar[r] = min_an[row];
    }

    if (MODE == 0) {
        #pragma unroll
        for (int r = 0; r < 8; ++r) {
            float vmin = __builtin_inff();
            #pragma unroll
            for (int c = 0; c < 8; ++c) {
                float g    = acc[c][r];
                float d2   = sqr[r] + sqc[c] - 2.0f * g;
                float dist = __builtin_sqrtf(fmaxf(d2, 0.0f) + 1e-12f);
                if (labc[c] != labr[r]) vmin = fminf(vmin, dist);
            }
            vmin = half16_min(vmin);
            if (l15 == 0) {
                // dist >= 0 and +inf: nonneg float order == uint order
                atomicMin((unsigned int*)(min_an + rowW + r), __float_as_uint(vmin));
            }
        }
    } else {
        float lsum = 0.0f;
        unsigned int cnt = 0;
        #pragma unroll
        for (int r = 0; r < 8; ++r) {
            int  row   = rowW + r;
            bool rowOk = labr[r] > 0;
            #pragma unroll
            for (int c = 0; c < 8; ++c) {
                int   col  = colBase + 16 * c + l15;
                float g    = acc[c][r];
                float d2   = sqr[r] + sqc[c] - 2.0f * g;
                float dist = __builtin_sqrtf(fmaxf(d2, 0.0f) + 1e-12f);
                float v    = dist + TMARGIN - mar[r];
                bool ok = rowOk && (labc[c] == labr[r]) && (row < col) && (v > 0.0f);
                lsum += ok ? v : 0.0f;
                cnt  += ok ? 1u : 0u;
            }
        }
        #pragma unroll
        for (int off = 1; off < 32; off <<= 1) {
            lsum += __shfl_xor(lsum, off, 32);
            cnt  += (unsigned int)__shfl_xor((int)cnt, off, 32);
        }
        if (lane == 0) {
            atomicAdd(acc_out, lsum);
            atomicAdd((unsigned int*)(acc_out + 1), cnt);
        }
    }
}

__global__ void finalize_kernel(const float* __restrict__ acc, float* __restrict__ out) {
    float        s = acc[0];
    unsigned int c = ((const unsigned int*)acc)[1];
    out[0] = s / (float)c;   // matches reference (nan if count == 0)
    out[1] = (float)c;
}

extern "C" void kernel_launch(void* const* d_in, const int* in_sizes, int n_in,
                              void* d_out, int out_size, void* d_ws, size_t ws_size,
                              hipStream_t stream) {
    const float* emb    = (const float*)d_in[0];
    const int*   labels = (const int*)d_in[1];
    float* ws     = (float*)d_ws;
    float* sq     = ws;            // [BSZ]
    float* min_an = ws + BSZ;      // [BSZ]
    float* acc    = ws + 2 * BSZ;  // [0]=loss sum (f32), [1]=count (u32 bits)

    sq_init_kernel<<<BSZ / 8, 256, 0, stream>>>(emb, sq, min_an, acc);

    dim3 grid(BSZ / TILE, BSZ / TILE);
    tile_kernel<0><<<grid, 256, 0, stream>>>(emb, labels, sq, min_an, acc);
    tile_kernel<1><<<grid, 256, 0, stream>>>(emb, labels, sq, min_an, acc);

    finalize_kernel<<<1, 1, 0, stream>>>(acc, (float*)d_out);
}